// EdgeClassifier_v1_35777077576523
// MI455X (gfx1250) — compile-verified
//
#include <hip/hip_runtime.h>
#include <math.h>

typedef __attribute__((ext_vector_type(2))) float v2f;
typedef __attribute__((ext_vector_type(8))) float v8f;

#define EMBED_DIM 128
#define ATTR_DIM  16
#define IN_DIM    272
#define HID       32
#define HSTRIDE   34   // even row stride for per-wave h tile: b64-aligned A loads, conflict-free
#define WAVES     8
#define EPB       128  // edges per block (8 waves x 16 edges)

#define WMMA_F32(a, b, c) \
    __builtin_amdgcn_wmma_f32_16x16x4_f32(false, (a), false, (b), (short)0, (c), false, false)

// Pack weight element W[kk][n] (row-major 32 cols) into the exact WMMA B-fragment
// layout: [kstep][colhalf][khalf][lane] as float2, so a lane's fragment is ONE
// aligned ds_load_b64 and a full-wave access covers all 64 LDS banks exactly once.
__device__ __forceinline__ int pack_idx(int kk, int n) {
    const int kstep = kk >> 2;
    const int khalf = (kk >> 1) & 1;
    const int j     = kk & 1;
    const int c     = n >> 4;
    const int ln    = n & 15;
    return ((((kstep << 2) + (c << 1) + khalf) << 4) + ln) * 2 + j;
}

__launch_bounds__(256)
__global__ void edge_mlp_wmma_f32(
    const float* __restrict__ emb,   // [N_NODES, 128]
    const float* __restrict__ attr,  // [E, 16]
    const int*   __restrict__ eidx,  // [2, E]
    const float* __restrict__ W1, const float* __restrict__ b1,  // [272,32],[32]
    const float* __restrict__ W2, const float* __restrict__ b2,  // [32,32],[32]
    const float* __restrict__ W3, const float* __restrict__ b3,  // [32,32],[32]
    const float* __restrict__ W4, const float* __restrict__ b4,  // [32,1],[1]
    float* __restrict__ out, int E)
{
    __shared__ __align__(16) float W1p[IN_DIM * HID];  // packed B fragments
    __shared__ __align__(16) float W2p[HID * HID];
    __shared__ __align__(16) float W3p[HID * HID];
    __shared__ float W4s[HID];
    __shared__ float b1s[HID], b2s[HID], b3s[HID];
    __shared__ __align__(16) float hbuf[WAVES * 16 * HSTRIDE];

    const int tid = threadIdx.x;

    // Cooperative stage of all weights into LDS in packed B-fragment order.
    for (int t = tid; t < IN_DIM * HID; t += 256)
        W1p[pack_idx(t >> 5, t & 31)] = W1[t];
    for (int t = tid; t < HID * HID; t += 256)
        W2p[pack_idx(t >> 5, t & 31)] = W2[t];
    for (int t = tid; t < HID * HID; t += 256)
        W3p[pack_idx(t >> 5, t & 31)] = W3[t];
    if (tid < HID) {
        W4s[tid] = W4[tid];
        b1s[tid] = b1[tid];
        b2s[tid] = b2[tid];
        b3s[tid] = b3[tid];
    }
    __syncthreads();

    const int wave = tid >> 5;
    const int lane = tid & 31;
    const int ln   = lane & 15;   // row within 16-edge tile / column within N-half
    const int half = lane >> 4;   // K-half selector for A/B fragments

    float* hs = &hbuf[wave * 16 * HSTRIDE];

    const int base = blockIdx.x * EPB + wave * 16;
    const int e    = base + ln;
    const int ec   = (e < E) ? e : (E - 1);   // clamp tail; store is predicated later

    const int srcn = eidx[ec];
    const int dstn = eidx[E + ec];

    // A-fragment source pointers: lane (ln,half) supplies x[ln][4*ks + 2*half .. +1]
    const float* __restrict__ srcp = emb  + (size_t)srcn * EMBED_DIM + half * 2;
    const float* __restrict__ dstp = emb  + (size_t)dstn * EMBED_DIM + half * 2;
    const float* __restrict__ atp  = attr + (size_t)ec   * ATTR_DIM  + half * 2;

    v8f acc0 = {};  // C tile, columns 0..15
    v8f acc1 = {};  // C tile, columns 16..31

    // ---------------- Layer 1: x[16x272] @ W1[272x32] ----------------
    // k-steps over src embedding (global kstep = ks)
    #pragma unroll 4
    for (int ks = 0; ks < EMBED_DIM / 4; ++ks) {
        v2f a = *(const v2f*)(srcp + ks * 4);
        const float* bp = &W1p[((ks * 4 + half) * 16 + ln) * 2];
        v2f bb0 = *(const v2f*)bp;          // colhalf 0
        v2f bb1 = *(const v2f*)(bp + 64);   // colhalf 1 (+2 groups of 16 float2)
        acc0 = WMMA_F32(a, bb0, acc0);
        acc1 = WMMA_F32(a, bb1, acc1);
    }
    // k-steps over dst embedding (global kstep = 32 + ks)
    #pragma unroll 4
    for (int ks = 0; ks < EMBED_DIM / 4; ++ks) {
        v2f a = *(const v2f*)(dstp + ks * 4);
        const float* bp = &W1p[(((ks + 32) * 4 + half) * 16 + ln) * 2];
        v2f bb0 = *(const v2f*)bp;
        v2f bb1 = *(const v2f*)(bp + 64);
        acc0 = WMMA_F32(a, bb0, acc0);
        acc1 = WMMA_F32(a, bb1, acc1);
    }
    // k-steps over edge attributes (global kstep = 64 + ks)
    #pragma unroll
    for (int ks = 0; ks < ATTR_DIM / 4; ++ks) {
        v2f a = *(const v2f*)(atp + ks * 4);
        const float* bp = &W1p[(((ks + 64) * 4 + half) * 16 + ln) * 2];
        v2f bb0 = *(const v2f*)bp;
        v2f bb1 = *(const v2f*)(bp + 64);
        acc0 = WMMA_F32(a, bb0, acc0);
        acc1 = WMMA_F32(a, bb1, acc1);
    }

    // bias + ReLU in C layout (VGPR r -> row r + 8*half, lane -> column), spill to LDS
    #pragma unroll
    for (int r = 0; r < 8; ++r) {
        const int row = r + half * 8;
        hs[row * HSTRIDE + ln]      = fmaxf(acc0[r] + b1s[ln],      0.0f);
        hs[row * HSTRIDE + 16 + ln] = fmaxf(acc1[r] + b1s[16 + ln], 0.0f);
    }
    asm volatile("s_wait_dscnt 0" ::: "memory");

    // ---------------- Layer 2: h1[16x32] @ W2[32x32] ----------------
    v8f d0 = {}, d1 = {};
    #pragma unroll
    for (int ks = 0; ks < HID / 4; ++ks) {
        v2f a = *(const v2f*)(&hs[ln * HSTRIDE + ks * 4 + half * 2]);
        const float* bp = &W2p[((ks * 4 + half) * 16 + ln) * 2];
        v2f bb0 = *(const v2f*)bp;
        v2f bb1 = *(const v2f*)(bp + 64);
        d0 = WMMA_F32(a, bb0, d0);
        d1 = WMMA_F32(a, bb1, d1);
    }
    asm volatile("s_wait_dscnt 0" ::: "memory");
    #pragma unroll
    for (int r = 0; r < 8; ++r) {
        const int row = r + half * 8;
        hs[row * HSTRIDE + ln]      = fmaxf(d0[r] + b2s[ln],      0.0f);
        hs[row * HSTRIDE + 16 + ln] = fmaxf(d1[r] + b2s[16 + ln], 0.0f);
    }
    asm volatile("s_wait_dscnt 0" ::: "memory");

    // ---------------- Layer 3: h2[16x32] @ W3[32x32] ----------------
    v8f g0 = {}, g1 = {};
    #pragma unroll
    for (int ks = 0; ks < HID / 4; ++ks) {
        v2f a = *(const v2f*)(&hs[ln * HSTRIDE + ks * 4 + half * 2]);
        const float* bp = &W3p[((ks * 4 + half) * 16 + ln) * 2];
        v2f bb0 = *(const v2f*)bp;
        v2f bb1 = *(const v2f*)(bp + 64);
        g0 = WMMA_F32(a, bb0, g0);
        g1 = WMMA_F32(a, bb1, g1);
    }
    asm volatile("s_wait_dscnt 0" ::: "memory");
    #pragma unroll
    for (int r = 0; r < 8; ++r) {
        const int row = r + half * 8;
        hs[row * HSTRIDE + ln]      = fmaxf(g0[r] + b3s[ln],      0.0f);
        hs[row * HSTRIDE + 16 + ln] = fmaxf(g1[r] + b3s[16 + ln], 0.0f);
    }
    asm volatile("s_wait_dscnt 0" ::: "memory");

    // ---------------- Layer 4: h3[16x32] @ W4[32x1] + sigmoid ----------------
    // lane (ln,half) reduces 16 of the 32 columns of row ln; combine halves via shfl.
    float partial = 0.0f;
    #pragma unroll
    for (int j = 0; j < 16; ++j)
        partial += hs[ln * HSTRIDE + half * 16 + j] * W4s[half * 16 + j];
    partial += __shfl_xor(partial, 16, 32);

    const float z = partial + b4[0];
    const float y = 1.0f / (1.0f + __expf(-z));
    if (half == 0 && e < E)
        out[e] = y;
}

extern "C" void kernel_launch(void* const* d_in, const int* in_sizes, int n_in,
                              void* d_out, int out_size, void* d_ws, size_t ws_size,
                              hipStream_t stream) {
    (void)n_in; (void)out_size; (void)d_ws; (void)ws_size;
    const float* emb  = (const float*)d_in[0];
    const float* attr = (const float*)d_in[1];
    const int*   eidx = (const int*)  d_in[2];
    const float* W1   = (const float*)d_in[3];
    const float* b1   = (const float*)d_in[4];
    const float* W2   = (const float*)d_in[5];
    const float* b2   = (const float*)d_in[6];
    const float* W3   = (const float*)d_in[7];
    const float* b3   = (const float*)d_in[8];
    const float* W4   = (const float*)d_in[9];
    const float* b4   = (const float*)d_in[10];
    float* out = (float*)d_out;

    const int E = in_sizes[2] / 2;  // edge_index is [2, E]
    const int blocks = (E + EPB - 1) / EPB;
    edge_mlp_wmma_f32<<<blocks, 256, 0, stream>>>(
        emb, attr, eidx, W1, b1, W2, b2, W3, b3, W4, b4, out, E);
}